// NLPD_Loss_12360915877964
// MI455X (gfx1250) — compile-verified
//
#include <hip/hip_runtime.h>
#include <stdint.h>

// NLPD loss: B=16, 512x512, 7 levels. All per-level sizes are even and >=8,
// so odd==0 everywhere and every level's H=W is a multiple of 16 (levels 0..5).

#define INV_GAMMA 0.3846153846153846f  /* 1/2.6 */
#define SIG0c 0.17f
#define SIG1c 4.86f

typedef float v2f __attribute__((ext_vector_type(2)));
typedef float v8f __attribute__((ext_vector_type(8)));

__device__ __forceinline__ int clampi(int v, int lo, int hi) {
  return v < lo ? lo : (v > hi ? hi : v);
}

// ---------------------------------------------------------------------------
// Wave32 sum reduction through V_WMMA_F32_16X16X4_F32.
// A: VGPR0 = s (A[m][0]=s_m for lanes 0-15, A[m][2]=s_{m+16} for lanes 16-31),
//    VGPR1 = 0 (kills K=1,3).  B = all ones (64 entries -> layout irrelevant).
// D[m][n] = s_m + s_{m+16}.  Sum of the 8 D VGPRs in lane n<16 gives
// sum_{m=0..7}, lanes>=16 give sum_{m=8..15}; xor-16 combine -> full wave sum.
// Requires EXEC all ones -> only called from fully-active blocks.
// ---------------------------------------------------------------------------
__device__ __forceinline__ float wave_sum32(float s) {
#if defined(__gfx1250__) && __has_builtin(__builtin_amdgcn_wmma_f32_16x16x4_f32)
  v2f a; a[0] = s;    a[1] = 0.0f;
  v2f o; o[0] = 1.0f; o[1] = 1.0f;
  v8f c = {};
  v8f d = __builtin_amdgcn_wmma_f32_16x16x4_f32(false, a, false, o,
                                                (short)0, c, false, false);
  float t = d[0] + d[1] + d[2] + d[3] + d[4] + d[5] + d[6] + d[7];
  return t + __shfl_xor(t, 16, 32);
#else
  float v = s;
#pragma unroll
  for (int k = 16; k > 0; k >>= 1) v += __shfl_xor(v, k, 32);
  return v;
#endif
}

// ---------------------------------------------------------------------------
// 1) gamma: out[c=b] = h^(1/g), out[c=16+b] = l^(1/g). Layout [32][512][512].
// ---------------------------------------------------------------------------
__global__ __launch_bounds__(256) void k_gamma(const float* __restrict__ h,
                                               const float* __restrict__ l,
                                               float* __restrict__ out, int n) {
  int i = blockIdx.x * 256 + threadIdx.x;
  if (i >= n) return;
  out[i]     = powf(h[i], INV_GAMMA);
  out[n + i] = powf(l[i], INV_GAMMA);
}

// ---------------------------------------------------------------------------
// 2) down: g[c][m][n] = sum_{u,v} F1[u][v] * in[c][clamp(2m+u-2)][clamp(2n+v-2)]
// ---------------------------------------------------------------------------
__global__ __launch_bounds__(256) void k_down(const float* __restrict__ in,
                                              float* __restrict__ out,
                                              const float* __restrict__ F1,
                                              int S) {
  int S2 = S >> 1;
  int total = 32 * S2 * S2;
  int i = blockIdx.x * 256 + threadIdx.x;
  if (i >= total) return;
  int n = i % S2, m = (i / S2) % S2, c = i / (S2 * S2);
  const float* src = in + (size_t)c * S * S;
  float a = 0.f;
#pragma unroll
  for (int u = 0; u < 5; u++) {
    int iy = clampi(2 * m + u - 2, 0, S - 1);
#pragma unroll
    for (int v = 0; v < 5; v++) {
      int ix = clampi(2 * n + v - 2, 0, S - 1);
      a += F1[u * 5 + v] * src[(size_t)iy * S + ix];
    }
  }
  out[i] = a;
}

// ---------------------------------------------------------------------------
// 3) lap = o - up(g):  up[y][x] = 4*sum_{u==y%2, v==x%2 (mod 2)}
//          F1[u][v] * g[clamp((y+u)/2-1)][clamp((x+v)/2-1)]
// ---------------------------------------------------------------------------
__global__ __launch_bounds__(256) void k_lap(const float* __restrict__ o,
                                             const float* __restrict__ g,
                                             const float* __restrict__ F1,
                                             float* __restrict__ lap, int S) {
  int S2 = S >> 1;
  int total = 32 * S * S;
  int i = blockIdx.x * 256 + threadIdx.x;
  if (i >= total) return;
  int x = i % S, y = (i / S) % S, c = i / (S * S);
  const float* gp = g + (size_t)c * S2 * S2;
  int ys = y & 1, xs = x & 1;
  int nu = 3 - ys, nv = 3 - xs;
  float a = 0.f;
  for (int uu = 0; uu < nu; uu++) {
    int u = ys + 2 * uu;
    int gy = clampi((y + u) / 2 - 1, 0, S2 - 1);
    for (int vv = 0; vv < nv; vv++) {
      int v = xs + 2 * vv;
      int gx = clampi((x + v) / 2 - 1, 0, S2 - 1);
      a += F1[u * 5 + v] * gp[(size_t)gy * S2 + gx];
    }
  }
  lap[i] = o[i] - 4.f * a;
}

// ---------------------------------------------------------------------------
// 4) norm+diff per 16x16 tile of one batch image pair. Stages the 20x20
//    replicate-clamped halo tiles of lap_h and lap_l into LDS with CDNA5
//    async global->LDS loads, waits on ASYNCcnt, then computes
//    d = (lap_h/den_h - lap_l/den_l)^2 and reduces to one partial per tile.
// ---------------------------------------------------------------------------
__global__ __launch_bounds__(256) void k_norm(const float* __restrict__ lap,
                                              const float* __restrict__ f0,
                                              float* __restrict__ part,
                                              int S, int ntiles) {
  __shared__ float sh[800];   // [2][20][20]
  __shared__ float red[8];
  int t = threadIdx.x;
  int b = blockIdx.z;
  int y0 = blockIdx.y * 16, x0 = blockIdx.x * 16;
  size_t plane = (size_t)S * S;
  const float* baseh = lap + (size_t)b * plane;
  const float* basel = lap + (size_t)(b + 16) * plane;

  // Async staging: 800 dwords, 256 lanes -> 4 rounds.
#pragma unroll
  for (int r = 0; r < 4; r++) {
    int idx = t + r * 256;
    if (idx < 800) {
      int ch  = idx / 400;
      int rem = idx - ch * 400;
      int p = rem / 20, q = rem - p * 20;
      int gy = clampi(y0 - 2 + p, 0, S - 1);
      int gx = clampi(x0 - 2 + q, 0, S - 1);
      const float* src = (ch ? basel : baseh) + (size_t)gy * S + gx;
      unsigned lds_off = (unsigned)(uintptr_t)(&sh[idx]);  // low 32 bits = LDS offset
      unsigned long long ga = (unsigned long long)(uintptr_t)src;
      asm volatile("global_load_async_to_lds_b32 %0, %1, off"
                   :: "v"(lds_off), "v"(ga) : "memory");
    }
  }
  asm volatile("s_wait_asynccnt 0" ::: "memory");
  __syncthreads();

  int tx = t & 15, ty = t >> 4;
  float denh = SIG0c, denl = SIG0c;
#pragma unroll
  for (int u = 0; u < 5; u++)
#pragma unroll
    for (int v = 0; v < 5; v++) {
      float w = f0[u * 5 + v];
      denh += w * fabsf(sh[(ty + u) * 20 + tx + v]);
      denl += w * fabsf(sh[400 + (ty + u) * 20 + tx + v]);
    }
  float lh = sh[(ty + 2) * 20 + tx + 2];
  float ll = sh[400 + (ty + 2) * 20 + tx + 2];
  float d = lh / denh - ll / denl;
  float s = d * d;                       // |x|^2.0

  float wsum = wave_sum32(s);            // EXEC all ones here
  int lane = t & 31, wid = t >> 5;
  if (lane == 0) red[wid] = wsum;
  __syncthreads();
  if (t == 0) {
    float tot = 0.f;
#pragma unroll
    for (int w = 0; w < 8; w++) tot += red[w];
    part[(size_t)b * ntiles + blockIdx.y * gridDim.x + blockIdx.x] = tot;
  }
}

// 5) deterministic tile-partials -> acc[b*7+level]
__global__ __launch_bounds__(256) void k_reduce(const float* __restrict__ part,
                                                float* __restrict__ acc,
                                                int ntiles, int level) {
  __shared__ float red[8];
  int b = blockIdx.x, t = threadIdx.x;
  float s = 0.f;
  for (int i = t; i < ntiles; i += 256) s += part[(size_t)b * ntiles + i];
  float wsum = wave_sum32(s);
  int lane = t & 31, wid = t >> 5;
  if (lane == 0) red[wid] = wsum;
  __syncthreads();
  if (t == 0) {
    float tot = 0.f;
#pragma unroll
    for (int w = 0; w < 8; w++) tot += red[w];
    acc[b * 7 + level] = tot;
  }
}

// 6) last level (8x8): filt1 is a center delta -> den = |x| + SIG1.
__global__ __launch_bounds__(64) void k_last(const float* __restrict__ g6,
                                             float* __restrict__ acc) {
  __shared__ float red[2];
  int b = blockIdx.x, t = threadIdx.x;
  float vh = g6[(size_t)b * 64 + t];
  float vl = g6[(size_t)(b + 16) * 64 + t];
  float nh = vh / (fabsf(vh) + SIG1c);
  float nl = vl / (fabsf(vl) + SIG1c);
  float d = nh - nl;
  float wsum = wave_sum32(d * d);
  int lane = t & 31, wid = t >> 5;
  if (lane == 0) red[wid] = wsum;
  __syncthreads();
  if (t == 0) acc[b * 7 + 6] = red[0] + red[1];
}

// 7) scalar finalize: dis=(sum/N)^{0.3}; loss_b=(mean_i dis)^{1/0.6}; mean_b.
__global__ void k_final(const float* __restrict__ acc, float* __restrict__ out) {
  float tot = 0.f;
  for (int b = 0; b < 16; b++) {
    float m = 0.f;
    int S = 512;
    for (int i = 0; i < 7; i++) {
      m += powf(acc[b * 7 + i] / ((float)S * (float)S), 0.3f);
      S >>= 1;
    }
    tot += powf(m / 7.f, 1.f / 0.6f);
  }
  *out = tot / 16.f;
}

// ---------------------------------------------------------------------------
extern "C" void kernel_launch(void* const* d_in, const int* in_sizes, int n_in,
                              void* d_out, int out_size, void* d_ws, size_t ws_size,
                              hipStream_t stream) {
  (void)in_sizes; (void)n_in; (void)out_size; (void)ws_size;
  const float* h_img = (const float*)d_in[0];
  const float* l_img = (const float*)d_in[1];
  const float* F1    = (const float*)d_in[2];
  const float* f0    = (const float*)d_in[3];
  // d_in[4]=filt1 (delta, folded analytically), d_in[5]=n_lev (fixed 7)

  // Workspace layout (floats):
  //   buf levels 0..6: [32][S][S] for S = 512,256,...,8
  //   lap scratch:     [32][512][512]  (reused per level)
  //   acc:             [16][7]
  //   part:            [16][1024] tile partials
  float* buf = (float*)d_ws;
  size_t off[8]; off[0] = 0;
  {
    int S = 512;
    for (int i = 0; i < 7; i++) { off[i + 1] = off[i] + (size_t)32 * S * S; S >>= 1; }
  }
  float* lapbuf = buf + off[7];
  float* acc    = lapbuf + (size_t)32 * 512 * 512;
  float* part   = acc + 16 * 7;

  const int n = 16 * 512 * 512;
  k_gamma<<<(n + 255) / 256, 256, 0, stream>>>(h_img, l_img, buf, n);

  int S = 512;
  for (int lev = 0; lev < 6; lev++) {
    int S2 = S >> 1;
    int tdown = 32 * S2 * S2;
    k_down<<<(tdown + 255) / 256, 256, 0, stream>>>(buf + off[lev], buf + off[lev + 1], F1, S);
    int tlap = 32 * S * S;
    k_lap<<<(tlap + 255) / 256, 256, 0, stream>>>(buf + off[lev], buf + off[lev + 1], F1, lapbuf, S);
    int tiles1d = S / 16;
    int ntiles = tiles1d * tiles1d;
    dim3 grid(tiles1d, tiles1d, 16);
    k_norm<<<grid, 256, 0, stream>>>(lapbuf, f0, part, S, ntiles);
    k_reduce<<<16, 256, 0, stream>>>(part, acc, ntiles, lev);
    S = S2;
  }
  k_last<<<16, 64, 0, stream>>>(buf + off[6], acc);
  k_final<<<1, 1, 0, stream>>>(acc, (float*)d_out);
}